// GraphVolterraLayer_85203561218642
// MI455X (gfx1250) — compile-verified
//
#include <hip/hip_runtime.h>

// GraphVolterraLayer on MI455X (gfx1250).
// All heavy contractions run on V_WMMA_F32_16X16X4_F32 (f32 in/out, full
// precision). One wave (32 lanes) computes one 16x16 C tile; strides encode
// all transposes. Problem is L2-resident (~20MB << 192MB); FLOPs ~0.35G.
//
// Sizes: N_S=64, N_T=32, N_EFF=2048, K_T1=K_T2=8, BATCH=16.
// Workspace requirement: 4,901,120 floats = ~19.6 MB.

typedef __attribute__((ext_vector_type(2))) float v2f;
typedef __attribute__((ext_vector_type(8))) float v8f;

#define NS 64
#define NT 32
#define NEFF 2048
#define BATCH 16

// ---------------------------------------------------------------------------
// Generic batched WMMA GEMM (f32):
//   C[b1,b2][m,n] = sum_k A[b1,b2][m,k] * B[b1,b2][k,n]
// Element addressing via strides (all in floats):
//   A: b1*sAb1 + b2*sAb2 + m*sAm + k*sAk
//   B: b1*sBb1 + b2*sBb2 + k*sBk + n*sBn
//   C: b1*sCb1 + b2*sCb2 + m*sCm + n        (n stride fixed at 1)
// Grid: x = tilesM*tilesN, y = b1, z = b2. Block: 32 threads (one wave).
// Requires M,N multiples of 16 and K multiple of 4. Full wave active ->
// EXEC all-ones as WMMA demands.
// ---------------------------------------------------------------------------
__global__ __launch_bounds__(32)
void wmma_gemm_f32(const float* __restrict__ A, const float* __restrict__ B,
                   float* __restrict__ C, int K,
                   int sAb1, int sAb2, int sAm, int sAk,
                   int sBb1, int sBb2, int sBk, int sBn,
                   int sCb1, int sCb2, int sCm, int tilesN)
{
    const int lane = threadIdx.x;            // 0..31
    const int tile = blockIdx.x;
    const int tm = (tile / tilesN) * 16;
    const int tn = (tile % tilesN) * 16;

    const float* Ab = A + (long)blockIdx.y * sAb1 + (long)blockIdx.z * sAb2;
    const float* Bb = B + (long)blockIdx.y * sBb1 + (long)blockIdx.z * sBb2;
    float*       Cb = C + (long)blockIdx.y * sCb1 + (long)blockIdx.z * sCb2;

    // ISA VGPR layout for 16x16x4 f32:
    //   A: lane holds A[m, kb] / A[m, kb+1], m = lane&15, kb = 2*(lane>>4)
    //   B: lane holds B[kb, n] / B[kb+1, n], n = lane&15
    //   C/D: VGPR v holds C[v + 8*(lane>>4), n]
    const int m  = tm + (lane & 15);
    const int n  = tn + (lane & 15);
    const int kb = (lane >> 4) << 1;

    v8f acc = {};
    for (int k0 = 0; k0 < K; k0 += 4) {
        v2f a, b;
        a.x = Ab[(long)m * sAm + (long)(k0 + kb)     * sAk];
        a.y = Ab[(long)m * sAm + (long)(k0 + kb + 1) * sAk];
        b.x = Bb[(long)(k0 + kb)     * sBk + (long)n * sBn];
        b.y = Bb[(long)(k0 + kb + 1) * sBk + (long)n * sBn];
        acc = __builtin_amdgcn_wmma_f32_16x16x4_f32(
            /*neg_a=*/false, a, /*neg_b=*/false, b,
            /*c_mod=*/(short)0, acc, /*reuse_a=*/false, /*reuse_b=*/false);
    }

    const int rbase = tm + ((lane >> 4) << 3);
#pragma unroll
    for (int v = 0; v < 8; ++v)
        Cb[(long)(rbase + v) * sCm + n] = acc[v];
}

// ---------------------------------------------------------------------------
// Small elementwise / helper kernels (latency-trivial)
// ---------------------------------------------------------------------------

// w[b,gk,tk] = fh1[gk,tk] * s_hat[b,gk,tk]       (total = 16*2048)
__global__ void k_mul_fh1(const float* __restrict__ shat,
                          const float* __restrict__ fh1,
                          float* __restrict__ w, int total)
{
    int i = blockIdx.x * blockDim.x + threadIdx.x;
    if (i < total) w[i] = shat[i] * fh1[i & (NEFF - 1)];
}

// C8[j,tl] = sum_m Hhat[j,m] * U_P_T[tl,m]    (8 x 32)
__global__ void k_c8(const float* __restrict__ Hh,
                     const float* __restrict__ UPT,
                     float* __restrict__ C8)
{
    int i = blockIdx.x * blockDim.x + threadIdx.x;   // 0..255
    if (i < 256) {
        int j = i >> 5, tl = i & 31;
        float s = 0.f;
#pragma unroll
        for (int m = 0; m < 8; ++m) s += Hh[j * 8 + m] * UPT[tl * 32 + m];
        C8[i] = s;
    }
}

// W1[t,j,x] = U_P_T[t,x] * U_P_T[x,j];  W2[t,j,x] = U_P_T[t,x] * C8[j,x]
__global__ void k_w12(const float* __restrict__ UPT,
                      const float* __restrict__ C8,
                      float* __restrict__ W1, float* __restrict__ W2)
{
    int i = blockIdx.x * blockDim.x + threadIdx.x;   // 0..8191
    if (i < 32 * 8 * 32) {
        int t = i >> 8, j = (i >> 5) & 7, x = i & 31;
        float u = UPT[t * 32 + x];
        W1[i] = u * UPT[x * 32 + j];
        W2[i] = u * C8[j * 32 + x];
    }
}

// In-place: Q[b,t,gk,gl] *= H_S[gk,gl]
__global__ void k_scale_hs(float* __restrict__ Q,
                           const float* __restrict__ HS, int total)
{
    int i = blockIdx.x * blockDim.x + threadIdx.x;
    if (i < total) Q[i] *= HS[i & (NS * NS - 1)];
}

// out[b,g,t] = h0[g,t] + z1[b,g,t] + sum_gk U_G[g,gk] * Z[b,t,gk,g]
__global__ void k_final(const float* __restrict__ h0,
                        const float* __restrict__ z1,
                        const float* __restrict__ Z,
                        const float* __restrict__ UG,
                        float* __restrict__ out)
{
    int i = blockIdx.x * blockDim.x + threadIdx.x;   // b*2048 + g*32 + t
    if (i >= BATCH * NEFF) return;
    int b = i >> 11, gt = i & (NEFF - 1);
    int g = gt >> 5, t = gt & 31;
    const float* Zp = Z + ((long)(b * NT + t) * NS) * NS + g;  // stride NS over gk
    float s = 0.f;
#pragma unroll 4
    for (int gk = 0; gk < NS; ++gk) s += UG[g * NS + gk] * Zp[(long)gk * NS];
    out[i] = h0[gt] + z1[i] + s;
}

// ---------------------------------------------------------------------------
// Host-side launch helper
// ---------------------------------------------------------------------------
static inline void gemm(hipStream_t st, const float* A, const float* B, float* C,
                        int M, int N, int K,
                        int sAb1, int sAb2, int sAm, int sAk,
                        int sBb1, int sBb2, int sBk, int sBn,
                        int sCb1, int sCb2, int sCm, int b1, int b2)
{
    int tilesN = N / 16;
    dim3 grid((M / 16) * tilesN, b1, b2);
    hipLaunchKernelGGL(wmma_gemm_f32, grid, dim3(32), 0, st,
                       A, B, C, K,
                       sAb1, sAb2, sAm, sAk,
                       sBb1, sBb2, sBk, sBn,
                       sCb1, sCb2, sCm, tilesN);
}

extern "C" void kernel_launch(void* const* d_in, const int* in_sizes, int n_in,
                              void* d_out, int out_size, void* d_ws, size_t ws_size,
                              hipStream_t stream)
{
    const float* s_in = (const float*)d_in[0];  // (16,2048)
    const float* UG   = (const float*)d_in[1];  // (64,64)
    const float* UPT  = (const float*)d_in[2];  // (32,32)
    const float* h0   = (const float*)d_in[3];  // (2048,)
    const float* h1c  = (const float*)d_in[4];  // (64,8)
    const float* HS   = (const float*)d_in[5];  // (64,64)
    const float* HhT  = (const float*)d_in[6];  // (8,8)
    float* out = (float*)d_out;

    float* ws = (float*)d_ws;
    float* tmpT = ws;                 // 32768   (b,gin,tk)
    float* shat = tmpT + 32768;       // 32768   (b,gk,tk)
    float* fh1  = shat + 32768;       // 2048    (gk,tk)
    float* w    = fh1  + 2048;        // 32768   (b,gk,tk)
    float* tmp2 = w    + 32768;       // 32768   (b,gk,t)
    float* z1   = tmp2 + 32768;       // 32768   (b,g,t)
    float* C8   = z1   + 32768;       // 256     (j,tl)
    float* W1   = C8   + 256;         // 8192    (t,j,tk)
    float* W2   = W1   + 8192;        // 8192    (t,j,tl)
    float* L    = W2   + 8192;        // 262144  (b,gk,t,j)
    float* R    = L    + 262144;      // 262144  (b,gl,t,j)
    float* Q    = R    + 262144;      // 2097152 (b,t,gk,gl)
    float* Z    = Q    + 2097152;     // 2097152 (b,t,gk,g)
    (void)in_sizes; (void)n_in; (void)out_size; (void)ws_size;

    // 1) tmpT[b,gin,tk] = sum_tin s[b,gin,tin] * U_P_T[tin,tk]
    gemm(stream, s_in, UPT, tmpT, BATCH * NS, NT, NT,
         0, 0, NT, 1,        0, 0, NT, 1,        0, 0, NT, 1, 1);

    // 2) s_hat[b,gk,tk] = sum_gin U_G[gin,gk] * tmpT[b,gin,tk]   (A = U_G^T)
    gemm(stream, UG, tmpT, shat, NS, NT, NS,
         0, 0, 1, NS,        NEFF, 0, NT, 1,     NEFF, 0, NT, BATCH, 1);

    // 3) fh1[gk,tk] = sum_j h1c[gk,j] * U_P_T[tk,j]   (B = Ub1^T via strides)
    gemm(stream, h1c, UPT, fh1, NS, NT, 8,
         0, 0, 8, 1,         0, 0, 1, NT,        0, 0, NT, 1, 1);

    // 4) w = fh1 (bcast over b) * s_hat
    hipLaunchKernelGGL(k_mul_fh1, dim3((BATCH * NEFF) / 256), dim3(256), 0, stream,
                       shat, fh1, w, BATCH * NEFF);

    // 5) tmp2[b,gk,t] = sum_tk w[b,gk,tk] * U_P_T[t,tk]
    gemm(stream, w, UPT, tmp2, BATCH * NS, NT, NT,
         0, 0, NT, 1,        0, 0, 1, NT,        0, 0, NT, 1, 1);

    // 6) z1[b,g,t] = sum_gk U_G[g,gk] * tmp2[b,gk,t]
    gemm(stream, UG, tmp2, z1, NS, NT, NS,
         0, 0, NS, 1,        NEFF, 0, NT, 1,     NEFF, 0, NT, BATCH, 1);

    // 7) C8 = Hhat2T @ Ub2^T
    hipLaunchKernelGGL(k_c8, dim3(1), dim3(256), 0, stream, HhT, UPT, C8);

    // 8) W1, W2
    hipLaunchKernelGGL(k_w12, dim3(8192 / 256), dim3(256), 0, stream, UPT, C8, W1, W2);

    // 9) L[b,gk,(t,j)] = sum_tk shat[b,gk,tk] * W1[t,j,tk]   (W1 addr = 32*n + k)
    gemm(stream, shat, W1, L, BATCH * NS, NT * 8, NT,
         0, 0, NT, 1,        0, 0, 1, NT,        0, 0, NT * 8, 1, 1);

    // 10) R likewise with W2
    gemm(stream, shat, W2, R, BATCH * NS, NT * 8, NT,
         0, 0, NT, 1,        0, 0, 1, NT,        0, 0, NT * 8, 1, 1);

    // 11) Q[b,t,gk,gl] = sum_j L[b,gk,t,j] * R[b,gl,t,j]   (batch dims b,t)
    gemm(stream, L, R, Q, NS, NS, 8,
         NS * NT * 8, 8, NT * 8, 1,
         NS * NT * 8, 8, 1, NT * 8,
         NT * NS * NS, NS * NS, NS, BATCH, NT);

    // 12) Q *= H_S (broadcast over b,t), in place
    hipLaunchKernelGGL(k_scale_hs, dim3((BATCH * NT * NS * NS) / 256), dim3(256),
                       0, stream, Q, HS, BATCH * NT * NS * NS);

    // 13) Z[b,t,gk,g] = sum_gl Q[b,t,gk,gl] * U_G[g,gl]   (B = U_G^T via strides)
    gemm(stream, Q, UG, Z, NS, NS, NS,
         NT * NS * NS, NS * NS, NS, 1,
         0, 0, 1, NS,
         NT * NS * NS, NS * NS, NS, BATCH, NT);

    // 14) out = h0 + z1 + diag_g( U_G @ Z )
    hipLaunchKernelGGL(k_final, dim3((BATCH * NEFF) / 256), dim3(256), 0, stream,
                       h0, z1, Z, UG, out);
}